// Temporal_Attention_12249246728619
// MI455X (gfx1250) — compile-verified
//
#include <hip/hip_runtime.h>
#include <math.h>

typedef float v2f __attribute__((ext_vector_type(2)));
typedef float v8f __attribute__((ext_vector_type(8)));

#define Hdim   56
#define Wdim   56
#define HW     3136       // 56*56
#define CIN    256
#define OUTC   256
#define C2     512        // 2*CIN in feature_map
#define KW     7
#define PAD    3
#define NTAPS  49
#define CG     32         // channels per group
#define NG     8
#define NB     2
#define TILE   8          // attention spatial tile
#define HALO   14         // TILE + 2*PAD
#define LDSTR  33         // HALO pixel channel stride (+1 pad vs 32 to spread banks)

// ---------------------------------------------------------------------------
// Kernel 1: channel projection GEMM  Y[b,o,hw] = sum_c W[o,c] * FM[b, coff+c, hw]
// One wave32 computes one 16(M=ochan) x 16(N=pixel) tile with
// V_WMMA_F32_16X16X4_F32, K-loop over 256 in steps of 4 (full fp32 precision).
// A 16x4 layout: lane L -> M = L&15, K-pair base = (L>>4)*2, V0/V1 = K+0/K+1.
// B 4x16 layout: lane L -> N = L&15, same K-pair striping.
// C/D: VGPR r -> M = r + 8*(L>>4), N = L&15.
// HW % 16 == 0, so a 16-pixel tile never crosses the batch boundary.
// ---------------------------------------------------------------------------
__global__ __launch_bounds__(256)
void proj_gemm_wmma(const float* __restrict__ Wm,
                    const float* __restrict__ FM,
                    float* __restrict__ Y,
                    int coff, int ntilesN)
{
    const int wave_id = (int)((blockIdx.x * blockDim.x + threadIdx.x) >> 5);
    const int lane    = (int)(threadIdx.x & 31);
    const int tileM   = wave_id / ntilesN;
    const int tileN   = wave_id % ntilesN;
    if (tileM >= OUTC / 16) return;          // uniform per-wave: EXEC stays all-1s

    const int o0 = tileM * 16;
    const int n0 = tileN * 16;
    const int l15 = lane & 15;
    const int kb  = (lane >> 4) << 1;        // 0 or 2: K-pair base for this half-wave

    // A operand: weight row (row-major -> contiguous 2-float reads)
    const float* wrow = Wm + (size_t)(o0 + l15) * CIN + kb;
    // B operand: feature-map column (channel stride = HW)
    const int n  = n0 + l15;
    const int bb = n / HW;
    const int hw = n % HW;
    const float* xcol = FM + ((size_t)(bb * C2 + coff + kb)) * HW + hw;

    v8f acc = {0.f, 0.f, 0.f, 0.f, 0.f, 0.f, 0.f, 0.f};
    #pragma unroll 4
    for (int c = 0; c < CIN; c += 4) {
        v2f a; a.x = wrow[c];               a.y = wrow[c + 1];
        v2f b; b.x = xcol[(size_t)c * HW];  b.y = xcol[(size_t)(c + 1) * HW];
        acc = __builtin_amdgcn_wmma_f32_16x16x4_f32(
                  false, a, false, b, (short)0, acc, false, false);
    }

    // Store D: M = r + 8*(lane>>4), N = lane&15
    const int row0 = (lane >> 4) * 8;
    float* ybase = Y + ((size_t)bb * OUTC) * HW + hw;
    #pragma unroll
    for (int r = 0; r < 8; ++r)
        ybase[(size_t)(o0 + row0 + r) * HW] = acc[r];
}

// ---------------------------------------------------------------------------
// Kernel 2: grouped 7x7 local attention with softmax.
// Block = 64 threads = one 8x8 pixel tile for one (batch, group).
// kk/vv halo tiles (14x14 x 32ch) staged in LDS (stride 33 to dodge bank
// conflicts). Bias trick: bias[c,t] depends only on tap-row (g<4: rel_h) or
// tap-col (g>=4: rel_w), so q.bias collapses to 7 dots per pixel (qb7).
// Out-of-range taps: k contribution is the bias dot only, v contribution 0
// (zero padding), exactly matching the reference's pad+unfold semantics.
// ---------------------------------------------------------------------------
__global__ __launch_bounds__(64)
void lsa_attention(const float* __restrict__ q,
                   const float* __restrict__ kk,
                   const float* __restrict__ vv,
                   const float* __restrict__ rel_h,
                   const float* __restrict__ rel_w,
                   float* __restrict__ out)
{
    __shared__ float ktile[HALO * HALO * LDSTR / 14];   // 14 * 33 = 462 floats
    __shared__ float vtile[HALO * HALO * LDSTR / 14];
    __shared__ float relS[CG * KW];

    const int g   = (int)blockIdx.y;
    const int bb  = (int)blockIdx.z;
    const int th0 = ((int)blockIdx.x / (Hdim / TILE)) * TILE;
    const int tw0 = ((int)blockIdx.x % (Wdim / TILE)) * TILE;
    const int tid = (int)threadIdx.x;

    // stage this group's 32 bias rows (K=7 each)
    const float* rel = (g < 4) ? (rel_h + (size_t)(g * CG) * KW)
                               : (rel_w + (size_t)(g * CG - 128) * KW);
    for (int i = tid; i < CG * KW; i += 64) relS[i] = rel[i];

    // stage kk/vv halo tiles (zero outside the image)
    const size_t chanBase = ((size_t)bb * OUTC + g * CG) * HW;
    for (int idx = tid; idx < 196 * CG; idx += 64) {
        const int cg = idx / 196;          // consecutive tid -> consecutive pixel
        const int p  = idx % 196;          //   -> mostly-contiguous global reads
        const int gy = th0 - PAD + p / HALO;
        const int gx = tw0 - PAD + p % HALO;
        float kv = 0.f, vval = 0.f;
        if (gy >= 0 && gy < Hdim && gx >= 0 && gx < Wdim) {
            const size_t off = chanBase + (size_t)cg * HW + (size_t)gy * Wdim + gx;
            kv   = kk[off];
            vval = vv[off];
        }
        ktile[p * LDSTR + cg] = kv;
        vtile[p * LDSTR + cg] = vval;
    }
    __syncthreads();

    const int ty = tid >> 3, tx = tid & 7;
    const int hw = (th0 + ty) * Wdim + (tw0 + tx);

    float qreg[CG];
    #pragma unroll
    for (int cg = 0; cg < CG; ++cg)
        qreg[cg] = q[chanBase + (size_t)cg * HW + hw];

    // 7 bias dots (exact: bias constant along the other tap axis)
    float qb7[KW];
    #pragma unroll
    for (int i = 0; i < KW; ++i) {
        float s = 0.f;
        #pragma unroll
        for (int cg = 0; cg < CG; ++cg) s += qreg[cg] * relS[cg * KW + i];
        qb7[i] = s;
    }

    float logits[NTAPS];
    float mx = -1e30f;
    for (int t = 0; t < NTAPS; ++t) {
        const int dy = t / KW, dx = t % KW;
        float s = qb7[(g < 4) ? dy : dx];
        const int gy = th0 + ty + dy - PAD;
        const int gx = tw0 + tx + dx - PAD;
        if (gy >= 0 && gy < Hdim && gx >= 0 && gx < Wdim) {
            const float* kt = &ktile[((ty + dy) * HALO + (tx + dx)) * LDSTR];
            float d = 0.f;
            #pragma unroll
            for (int cg = 0; cg < CG; ++cg) d += qreg[cg] * kt[cg];
            s += d;
        }
        logits[t] = s;
        mx = fmaxf(mx, s);
    }

    float sum = 0.f;
    #pragma unroll
    for (int t = 0; t < NTAPS; ++t) { logits[t] = __expf(logits[t] - mx); sum += logits[t]; }
    const float inv = 1.0f / sum;

    float accv[CG];
    #pragma unroll
    for (int cg = 0; cg < CG; ++cg) accv[cg] = 0.f;
    for (int t = 0; t < NTAPS; ++t) {
        const int dy = t / KW, dx = t % KW;
        const int gy = th0 + ty + dy - PAD;
        const int gx = tw0 + tx + dx - PAD;
        if (gy >= 0 && gy < Hdim && gx >= 0 && gx < Wdim) {
            const float wt = logits[t] * inv;
            const float* vt = &vtile[((ty + dy) * HALO + (tx + dx)) * LDSTR];
            #pragma unroll
            for (int cg = 0; cg < CG; ++cg) accv[cg] += wt * vt[cg];
        }
    }

    #pragma unroll
    for (int cg = 0; cg < CG; ++cg)
        out[chanBase + (size_t)cg * HW + hw] = accv[cg];
}

// ---------------------------------------------------------------------------
extern "C" void kernel_launch(void* const* d_in, const int* in_sizes, int n_in,
                              void* d_out, int out_size, void* d_ws, size_t ws_size,
                              hipStream_t stream) {
    const float* fm    = (const float*)d_in[0];   // (2, 512, 56, 56)
    const float* w_q   = (const float*)d_in[1];   // (256, 256)
    const float* w_k   = (const float*)d_in[2];
    const float* w_v   = (const float*)d_in[3];
    const float* rel_h = (const float*)d_in[4];   // (128, 7)
    const float* rel_w = (const float*)d_in[5];   // (128, 7)
    float* out = (float*)d_out;                   // (2, 256, 56, 56)

    const size_t planeSz = (size_t)NB * OUTC * HW;   // 1,605,632 floats
    float* qbuf = (float*)d_ws;
    float* kbuf = qbuf + planeSz;
    float* vbuf = kbuf + planeSz;

    const int ntilesN    = (NB * HW) / 16;           // 392 (HW%16==0: tiles stay in-batch)
    const int totalWaves = (OUTC / 16) * ntilesN;    // 6272 waves
    const int blocks     = (totalWaves * 32 + 255) / 256;

    // q from fm_t1 (channel offset 256); kk, vv from fm_t0 (offset 0)
    proj_gemm_wmma<<<blocks, 256, 0, stream>>>(w_q, fm, qbuf, CIN, ntilesN);
    proj_gemm_wmma<<<blocks, 256, 0, stream>>>(w_k, fm, kbuf, 0,   ntilesN);
    proj_gemm_wmma<<<blocks, 256, 0, stream>>>(w_v, fm, vbuf, 0,   ntilesN);

    dim3 grid((Hdim / TILE) * (Wdim / TILE), NG, NB);  // (49, 8, 2)
    lsa_attention<<<grid, 64, 0, stream>>>(qbuf, kbuf, vbuf, rel_h, rel_w, out);
}